// GAT_all_10960756540167
// MI455X (gfx1250) — compile-verified
//
#include <hip/hip_runtime.h>
#include <hip/hip_bf16.h>

// ---------------------------------------------------------------------------
// GAT forward for MI455X (gfx1250), fp32 WMMA 16x16x4 everywhere.
// N=4096, NFEAT=512, NHID=64, NHEADS=8, NREL=64, NCLASS=16
// ---------------------------------------------------------------------------

typedef __attribute__((ext_vector_type(2))) float v2f;
typedef __attribute__((ext_vector_type(8))) float v8f;

#define NN      4096
#define NFEAT   512
#define NHID    64
#define NHEADS  8
#define NREL    64
#define NCLASS  16
#define ALPHA   0.2f

__device__ __forceinline__ v8f wmma4(v2f a, v2f b, v8f c) {
    // V_WMMA_F32_16X16X4_F32 : D = A(16x4) * B(4x16) + C(16x16)
    return __builtin_amdgcn_wmma_f32_16x16x4_f32(
        false, a, false, b, (short)0, c, false, false);
}

__device__ __forceinline__ float lrelu(float x) { return x > 0.f ? x : ALPHA * x; }
__device__ __forceinline__ float eluf(float x)  { return x > 0.f ? x : (__expf(x) - 1.f); }

// ---------------------------------------------------------------------------
// Repack W_heads [8][512][64] -> Wcat [512][512] with col = h*64+c
// ---------------------------------------------------------------------------
__global__ void packW_kernel(const float* __restrict__ Wh, float* __restrict__ Wcat) {
    int idx = blockIdx.x * blockDim.x + threadIdx.x;
    if (idx >= NFEAT * (NHEADS * NHID)) return;
    int k = idx >> 9;          // row (512)
    int j = idx & 511;         // col (512)
    int h = j >> 6, c = j & 63;
    Wcat[idx] = Wh[h * (NFEAT * NHID) + k * NHID + c];
}

// ---------------------------------------------------------------------------
// adj (fp32 dense, 64MB) -> bitmask (2MB, L2-resident). One word per thread.
// ---------------------------------------------------------------------------
__global__ void mask_build_kernel(const float* __restrict__ adj, unsigned* __restrict__ mb) {
    int w = blockIdx.x * blockDim.x + threadIdx.x;       // 0 .. 4096*128-1
    if (w >= NN * (NN / 32)) return;
    const float* p = adj + (size_t)w * 32;
    unsigned m = 0;
#pragma unroll
    for (int b = 0; b < 32; ++b)
        if (p[b] > 0.f) m |= (1u << b);
    mb[w] = m;
}

// ---------------------------------------------------------------------------
// Masked row-max of f2 per head:  Mrow[h][i] = max_{j: mask(i,j)} f2[h][j]
// (softmax row max = lrelu(f1_i + Mrow_i) by monotonicity of leaky-relu)
// ---------------------------------------------------------------------------
__global__ __launch_bounds__(128) void rowmax_masked_kernel(
    const unsigned* __restrict__ mb, const float* __restrict__ f2,
    float* __restrict__ Mrow, int nheads) {
    __shared__ float red[128];
    int i = blockIdx.x, t = threadIdx.x;
    unsigned word = mb[i * (NN / 32) + t];
    float mx[NHEADS];
#pragma unroll
    for (int h = 0; h < NHEADS; ++h) mx[h] = -3.0e38f;
    unsigned w = word;
    while (w) {
        int b = __ffs((int)w) - 1;
        w &= (w - 1);
        int j = t * 32 + b;
        for (int h = 0; h < nheads; ++h)
            mx[h] = fmaxf(mx[h], f2[h * NN + j]);
    }
    for (int h = 0; h < nheads; ++h) {
        red[t] = mx[h];
        __syncthreads();
        for (int s = 64; s > 0; s >>= 1) {
            if (t < s) red[t] = fmaxf(red[t], red[t + s]);
            __syncthreads();
        }
        if (t == 0) Mrow[h * NN + i] = red[0];
        __syncthreads();
    }
}

// ---------------------------------------------------------------------------
// f1[h][i] = dot(H[i, h*fdim : (h+1)*fdim], a1[h]);  same for f2.
// nheads=8/fdim=64 (layer1) or nheads=1/fdim=512 (layer2). ldh = 512.
// ---------------------------------------------------------------------------
__global__ void fdot_kernel(const float* __restrict__ H,
                            const float* __restrict__ a1, const float* __restrict__ a2,
                            float* __restrict__ f1, float* __restrict__ f2,
                            int nheads, int fdim) {
    int idx = blockIdx.x * blockDim.x + threadIdx.x;
    if (idx >= NN * nheads) return;
    int h = idx >> 12;           // idx / 4096
    int i = idx & (NN - 1);
    const float* hp = H + (size_t)i * NFEAT + h * fdim;
    const float* a1p = a1 + h * fdim;
    const float* a2p = a2 + h * fdim;
    float s1 = 0.f, s2 = 0.f;
    for (int c = 0; c < fdim; ++c) {
        float v = hp[c];
        s1 += v * a1p[c];
        s2 += v * a2p[c];
    }
    f1[h * NN + i] = s1;
    f2[h * NN + i] = s2;
}

// ---------------------------------------------------------------------------
// Generic fp32 WMMA GEMM: C[M,N] = A[M,K] @ B[K,N]  (+epilogues)
//   epi=0: none   epi=1: +bias[col], elu   epi=2: +resid[row,col]
// Each wave owns a 16 x (16*NT) output tile: the A fragment is loaded once
// per k-step and reused across NT WMMAs (NT=4 for the 512-wide GEMMs).
// ---------------------------------------------------------------------------
template <int NT>
__global__ __launch_bounds__(128) void wmma_gemm_kernel(
    const float* __restrict__ A, const float* __restrict__ B, float* __restrict__ C,
    int M, int N, int K,
    const float* __restrict__ bias, const float* __restrict__ resid, int epi) {
    int lane = threadIdx.x & 31;
    int wid  = threadIdx.x >> 5;
    int tilesN = N / (16 * NT);
    int tile = blockIdx.x * 4 + wid;
    if (tile >= (M >> 4) * tilesN) return;
    int tm = tile / tilesN;
    int tn = tile - tm * tilesN;
    int li = lane & 15;
    int khalf = (lane >> 4) << 1;      // lanes 0-15 -> K {0,1}; 16-31 -> K {2,3}
    int row  = tm * 16 + li;
    int colb = tn * (16 * NT) + li;

    v8f acc[NT];
#pragma unroll
    for (int nt = 0; nt < NT; ++nt) acc[nt] = (v8f){0.f,0.f,0.f,0.f,0.f,0.f,0.f,0.f};

    const float* Ap = A + (size_t)row * K + khalf;   // a.x/a.y contiguous -> b64 load
    for (int k = 0; k < K; k += 4) {
        v2f a;
        a.x = Ap[k];
        a.y = Ap[k + 1];
        const float* Bp = B + (size_t)(k + khalf) * N + colb;
#pragma unroll
        for (int nt = 0; nt < NT; ++nt) {
            v2f b;
            b.x = Bp[nt * 16];
            b.y = Bp[nt * 16 + N];
            acc[nt] = wmma4(a, b, acc[nt]);
        }
    }

    int mo = (lane >> 4) << 3;         // lanes>=16 hold rows M=v+8
#pragma unroll
    for (int nt = 0; nt < NT; ++nt) {
#pragma unroll
        for (int v = 0; v < 8; ++v) {
            int m = v + mo;
            size_t o = (size_t)(tm * 16 + m) * N + colb + nt * 16;
            float val = acc[nt][v];
            if (epi == 1)      { val += bias[colb + nt * 16]; val = eluf(val); }
            else if (epi == 2) { val += resid[o]; }
            C[o] = val;
        }
    }
}

// ---------------------------------------------------------------------------
// Fused masked-softmax attention + aggregation.
//   OUT[i, cols] = softmax_j(mask, lrelu(f1_i + f2_j)) @ H[:, cols]
// Block = 8 waves, 32 rows (2 x 16-row subtiles) per block.
//   multihead=1: wave w == head w, cols w*64..w*64+63, per-head f1/f2/M, ELU.
//   multihead=0: single head, wave w covers cols w*64..w*64+63, no ELU.
// exp() values are generated directly in WMMA A-fragment layout; the softmax
// denominator is accumulated alongside and applied in the epilogue.
// ---------------------------------------------------------------------------
__global__ __launch_bounds__(256) void attn_agg_kernel(
    const float* __restrict__ H, float* __restrict__ OUT,
    const float* __restrict__ f1, const float* __restrict__ f2,
    const float* __restrict__ Mrow, const unsigned* __restrict__ mb,
    int multihead) {
    __shared__ float sS[8][2][16];

    int lane = threadIdx.x & 31;
    int wid  = threadIdx.x >> 5;
    int head = multihead ? wid : 0;
    int foff = head * NN;
    int colbase = wid * 64;
    int i0 = blockIdx.x * 32;
    int li = lane & 15;
    int khalf = (lane >> 4) << 1;

    int   irow[2];
    float f1v[2], mi[2];
#pragma unroll
    for (int r = 0; r < 2; ++r) {
        irow[r] = i0 + r * 16 + li;
        f1v[r]  = f1[foff + irow[r]];
        mi[r]   = lrelu(f1v[r] + Mrow[foff + irow[r]]);  // exact softmax row max
    }

    v8f acc[2][4];
#pragma unroll
    for (int r = 0; r < 2; ++r)
#pragma unroll
        for (int nt = 0; nt < 4; ++nt) acc[r][nt] = (v8f){0.f,0.f,0.f,0.f,0.f,0.f,0.f,0.f};
    float ssum[2] = {0.f, 0.f};

    for (int jt = 0; jt < NN / 16; ++jt) {
        // prefetch H rows 8 j-tiles ahead (L2-resident; warms near caches)
        if (jt + 8 < NN / 16) {
            int pr = jt * 16 + 128 + (lane >> 1);
            __builtin_prefetch(&H[(size_t)pr * NFEAT + colbase + (lane & 1) * 32], 0, 1);
        }
        unsigned mw[2];
        mw[0] = mb[irow[0] * (NN / 32) + (jt >> 1)] >> ((jt & 1) * 16);
        mw[1] = mb[irow[1] * (NN / 32) + (jt >> 1)] >> ((jt & 1) * 16);
        int jt16 = jt * 16;
#pragma unroll
        for (int kk = 0; kk < 4; ++kk) {
            int tb = kk * 4 + khalf;              // bit position inside 16-wide j tile
            int j0 = jt16 + tb;
            float f2v0 = f2[foff + j0];
            float f2v1 = f2[foff + j0 + 1];
            v2f afr[2];
#pragma unroll
            for (int r = 0; r < 2; ++r) {
                float e0 = lrelu(f1v[r] + f2v0);
                float e1 = lrelu(f1v[r] + f2v1);
                float p0 = ((mw[r] >> tb) & 1u)       ? __expf(e0 - mi[r]) : 0.f;
                float p1 = ((mw[r] >> (tb + 1)) & 1u) ? __expf(e1 - mi[r]) : 0.f;
                ssum[r] += p0 + p1;
                afr[r].x = p0;
                afr[r].y = p1;
            }
#pragma unroll
            for (int nt = 0; nt < 4; ++nt) {
                int col = colbase + nt * 16 + li;
                v2f b;
                b.x = H[(size_t)j0 * NFEAT + col];
                b.y = H[(size_t)(j0 + 1) * NFEAT + col];
                acc[0][nt] = wmma4(afr[0], b, acc[0][nt]);
                acc[1][nt] = wmma4(afr[1], b, acc[1][nt]);
            }
        }
    }

    // softmax denominators: lane li and li+16 hold complementary halves of row li
#pragma unroll
    for (int r = 0; r < 2; ++r) {
        float s = ssum[r] + __shfl_xor(ssum[r], 16, 32);
        if (lane < 16) sS[wid][r][li] = s;
    }
    __syncthreads();

    int mo = (lane >> 4) << 3;
#pragma unroll
    for (int r = 0; r < 2; ++r) {
#pragma unroll
        for (int nt = 0; nt < 4; ++nt) {
#pragma unroll
            for (int v = 0; v < 8; ++v) {
                int m = v + mo;
                float val = acc[r][nt][v] / sS[wid][r][m];
                if (multihead) val = eluf(val);
                OUT[(size_t)(i0 + r * 16 + m) * NFEAT + colbase + nt * 16 + li] = val;
            }
        }
    }
}

// ---------------------------------------------------------------------------
// Row-wise log_softmax over 16 classes -> final output
// ---------------------------------------------------------------------------
__global__ void logsoftmax16_kernel(const float* __restrict__ logits, float* __restrict__ out) {
    int i = blockIdx.x * blockDim.x + threadIdx.x;
    if (i >= NN) return;
    const float* p = logits + (size_t)i * NCLASS;
    float mx = p[0];
#pragma unroll
    for (int c = 1; c < NCLASS; ++c) mx = fmaxf(mx, p[c]);
    float s = 0.f;
#pragma unroll
    for (int c = 0; c < NCLASS; ++c) s += __expf(p[c] - mx);
    float ls = logf(s) + mx;
#pragma unroll
    for (int c = 0; c < NCLASS; ++c) out[(size_t)i * NCLASS + c] = p[c] - ls;
}

// ---------------------------------------------------------------------------
extern "C" void kernel_launch(void* const* d_in, const int* in_sizes, int n_in,
                              void* d_out, int out_size, void* d_ws, size_t ws_size,
                              hipStream_t stream) {
    const float* x       = (const float*)d_in[0];
    const float* rel     = (const float*)d_in[1];
    /* d_in[2] = rel_dict (unused by the math) */
    const float* adj     = (const float*)d_in[3];
    const float* adj_ad  = (const float*)d_in[4];
    const float* W_heads = (const float*)d_in[5];
    const float* a1h     = (const float*)d_in[6];
    const float* a2h     = (const float*)d_in[7];
    const float* W_out   = (const float*)d_in[8];
    const float* a1o     = (const float*)d_in[9];
    const float* a2o     = (const float*)d_in[10];
    const float* lin_W   = (const float*)d_in[11];
    const float* lin_b   = (const float*)d_in[12];
    float* out = (float*)d_out;

    char* ws = (char*)d_ws;
    size_t off = 0;
    auto alloc = [&](size_t bytes) -> void* {
        void* p = ws + off;
        off += (bytes + 255) & ~(size_t)255;
        return p;
    };
    float*    xr     = (float*)   alloc((size_t)NN * NFEAT * 4);
    float*    Wcat   = (float*)   alloc((size_t)NFEAT * NFEAT * 4);
    float*    H1     = (float*)   alloc((size_t)NN * NFEAT * 4);
    float*    f1h    = (float*)   alloc((size_t)NHEADS * NN * 4);
    float*    f2h    = (float*)   alloc((size_t)NHEADS * NN * 4);
    float*    M1     = (float*)   alloc((size_t)NHEADS * NN * 4);
    float*    hcat   = (float*)   alloc((size_t)NN * NFEAT * 4);
    float*    xr2    = (float*)   alloc((size_t)NN * NFEAT * 4);
    float*    H2     = (float*)   alloc((size_t)NN * NFEAT * 4);
    float*    f1b    = (float*)   alloc((size_t)NN * 4);
    float*    f2b    = (float*)   alloc((size_t)NN * 4);
    float*    M2     = (float*)   alloc((size_t)NN * 4);
    float*    out2   = (float*)   alloc((size_t)NN * NFEAT * 4);
    float*    logits = (float*)   alloc((size_t)NN * NCLASS * 4);
    unsigned* mb     = (unsigned*)alloc((size_t)NN * (NN / 32) * 4);

    // wide GEMMs: 16x64 per wave -> tiles = (M/16) * (N/64), 4 waves/block
    const int tilesWide = (NN / 16) * (NFEAT / 64);   // 2048

    // one-time prep
    packW_kernel<<<(NFEAT * NFEAT) / 256, 256, 0, stream>>>(W_heads, Wcat);
    mask_build_kernel<<<(NN * (NN / 32)) / 256, 256, 0, stream>>>(adj, mb);

    // ---- layer 1 ----
    // xr = x + adj_ad @ rel
    wmma_gemm_kernel<4><<<tilesWide / 4, 128, 0, stream>>>(adj_ad, rel, xr,
        NN, NFEAT, NREL, nullptr, x, 2);
    // H1 = xr @ Wcat    (all 8 heads as one GEMM)
    wmma_gemm_kernel<4><<<tilesWide / 4, 128, 0, stream>>>(xr, Wcat, H1,
        NN, NFEAT, NFEAT, nullptr, nullptr, 0);
    fdot_kernel<<<(NN * NHEADS) / 256, 256, 0, stream>>>(H1, a1h, a2h, f1h, f2h, NHEADS, NHID);
    rowmax_masked_kernel<<<NN, 128, 0, stream>>>(mb, f2h, M1, NHEADS);
    attn_agg_kernel<<<NN / 32, 256, 0, stream>>>(H1, hcat, f1h, f2h, M1, mb, 1);

    // ---- layer 2 ----
    // xr2 = hcat + adj_ad @ rel
    wmma_gemm_kernel<4><<<tilesWide / 4, 128, 0, stream>>>(adj_ad, rel, xr2,
        NN, NFEAT, NREL, nullptr, hcat, 2);
    // H2 = xr2 @ W_out
    wmma_gemm_kernel<4><<<tilesWide / 4, 128, 0, stream>>>(xr2, W_out, H2,
        NN, NFEAT, NFEAT, nullptr, nullptr, 0);
    fdot_kernel<<<NN / 256, 256, 0, stream>>>(H2, a1o, a2o, f1b, f2b, 1, NFEAT);
    rowmax_masked_kernel<<<NN, 128, 0, stream>>>(mb, f2b, M2, 1);
    attn_agg_kernel<<<NN / 32, 256, 0, stream>>>(H2, out2, f1b, f2b, M2, mb, 0);

    // ---- classifier ----
    // logits = elu(out2 @ lin_W + lin_b)   (N=16 -> narrow tile variant)
    {
        int tiles = (NN / 16) * (NCLASS / 16);       // 256
        wmma_gemm_kernel<1><<<tiles / 4, 128, 0, stream>>>(out2, lin_W, logits,
            NN, NCLASS, NFEAT, lin_b, nullptr, 1);
    }
    logsoftmax16_kernel<<<NN / 256, 256, 0, stream>>>(logits, out);
}